// FaceRecognitionAccuracy_23441931502276
// MI455X (gfx1250) — compile-verified
//
#include <hip/hip_runtime.h>
#include <cstdint>
#include <cstddef>

// Problem constants (from the reference).
#define NROWS 262144
#define DIM   128

// Tunables.
constexpr int WAVES_PER_BLOCK = 8;
constexpr int BLOCK           = 32 * WAVES_PER_BLOCK;          // 256 threads
constexpr int ROWS_PER_STAGE  = 8;                             // rows per wave per LDS buffer
constexpr int STAGES          = 8;                             // stages per wave
constexpr int ROWS_PER_WAVE   = ROWS_PER_STAGE * STAGES;       // 64
constexpr int NBLOCKS         = NROWS / (ROWS_PER_WAVE * WAVES_PER_BLOCK); // 512

constexpr int ROW_BYTES    = DIM * 4;                          // 512
constexpr int ROW_STRIDE_B = ROW_BYTES + 16;                   // 528: pad -> minimal LDS bank conflicts
constexpr int BUF_B        = ROWS_PER_STAGE * 2 * ROW_STRIDE_B; // 8448 (o1 rows then o2 rows)
constexpr int WAVE_B       = 2 * BUF_B;                        // 16896 (double buffer)
constexpr int SMEM_B       = WAVES_PER_BLOCK * WAVE_B;         // 135168

__global__ __launch_bounds__(BLOCK) void dist_iou_partial_kernel(
    const float* __restrict__ o1, const float* __restrict__ o2,
    const float* __restrict__ desire, float* __restrict__ ws)
{
  __shared__ __align__(16) char smem[SMEM_B];
  __shared__ float rA[BLOCK], rB[BLOCK], rC[BLOCK];

  const int tid  = threadIdx.x;
  const int wave = tid >> 5;
  const int lane = tid & 31;

  // 32-bit LDS byte address of this wave's staging region (low 32 bits of the
  // flat shared-aperture pointer are the LDS offset the HW uses).
  const unsigned ldsBase = (unsigned)(size_t)&smem[0] + (unsigned)(wave * WAVE_B);
  const unsigned smemOff = (unsigned)(wave * WAVE_B);  // same region, for ds reads

  const long long waveRow0 =
      ((long long)blockIdx.x * WAVES_PER_BLOCK + wave) * (long long)ROWS_PER_WAVE;

  const int r = lane & 7;   // row-in-stage owned by this lane (4 lanes/row)
  const int q = lane >> 3;  // quarter of the row this lane reduces

  float accA = 0.f;  // sum(actual)
  float accB = 0.f;  // sum(actual * desire)
  float accC = 0.f;  // sum(desire)

  // Issue one stage: 16 async DMA ops, each moving one contiguous 512B row
  // (lane L copies bytes [16L,16L+16)) into this wave's LDS buffer b.
  auto issue = [&](int s, int b) {
    // Ensure prior ds_load reads of the buffer being overwritten have drained.
    asm volatile("s_wait_dscnt 0" ::: "memory");
    const long long row0 = waveRow0 + (long long)s * ROWS_PER_STAGE;
    const unsigned dst0  = ldsBase + (unsigned)(b * BUF_B) + (unsigned)lane * 16u;
    const float* s1 = o1 + row0 * DIM + lane * 4;
    const float* s2 = o2 + row0 * DIM + lane * 4;
#pragma unroll
    for (int rr = 0; rr < ROWS_PER_STAGE; ++rr) {
      asm volatile("global_load_async_to_lds_b128 %0, %1, off"
                   :: "v"(dst0 + (unsigned)(rr * ROW_STRIDE_B)),
                      "v"(s1 + rr * DIM)
                   : "memory");
      asm volatile("global_load_async_to_lds_b128 %0, %1, off"
                   :: "v"(dst0 + (unsigned)((ROWS_PER_STAGE + rr) * ROW_STRIDE_B)),
                      "v"(s2 + rr * DIM)
                   : "memory");
    }
  };

  auto compute = [&](int s, int b) {
    const unsigned base = smemOff + (unsigned)(b * BUF_B);
    const unsigned offA = base + (unsigned)(r * ROW_STRIDE_B) + (unsigned)(q * 128);
    const unsigned offB = offA + (unsigned)(ROWS_PER_STAGE * ROW_STRIDE_B);
    float sum = 0.f;
#pragma unroll
    for (int k = 0; k < 8; ++k) {
      const float4 a = *(const float4*)&smem[offA + (unsigned)(k * 16)];
      const float4 c = *(const float4*)&smem[offB + (unsigned)(k * 16)];
      const float dx = a.x - c.x, dy = a.y - c.y;
      const float dz = a.z - c.z, dw = a.w - c.w;
      sum = fmaf(dx, dx, sum);
      sum = fmaf(dy, dy, sum);
      sum = fmaf(dz, dz, sum);
      sum = fmaf(dw, dw, sum);
    }
    // Combine the 4 lanes that share this row.
    sum += __shfl_xor(sum, 8, 32);
    sum += __shfl_xor(sum, 16, 32);
    if (q == 0) {
      const float dsr = desire[waveRow0 + (long long)s * ROWS_PER_STAGE + r];
      // dist > 0.625  <=>  dist^2 > 0.390625 (sqrt-free, monotone)
      const float act = (sum > 0.390625f) ? 1.f : 0.f;
      accA += act;
      accB += act * dsr;
      accC += dsr;
    }
  };

  // Double-buffered async pipeline.
  issue(0, 0);
#pragma unroll 1
  for (int s = 0; s < STAGES; ++s) {
    if (s + 1 < STAGES) {
      issue(s + 1, (s + 1) & 1);
      // 16 newer asyncs outstanding; in-order completion => stage s has landed.
      asm volatile("s_wait_asynccnt 16" ::: "memory");
    } else {
      asm volatile("s_wait_asynccnt 0" ::: "memory");
    }
    compute(s, s & 1);
  }

  // Deterministic block reduction (fixed tree; no float atomics).
  rA[tid] = accA; rB[tid] = accB; rC[tid] = accC;
  __syncthreads();
  for (int off = BLOCK / 2; off > 0; off >>= 1) {
    if (tid < off) {
      rA[tid] += rA[tid + off];
      rB[tid] += rB[tid + off];
      rC[tid] += rC[tid + off];
    }
    __syncthreads();
  }
  if (tid == 0) {
    ws[blockIdx.x]               = rA[0];
    ws[NBLOCKS + blockIdx.x]     = rB[0];
    ws[2 * NBLOCKS + blockIdx.x] = rC[0];
  }
}

__global__ __launch_bounds__(256) void dist_iou_finalize_kernel(
    const float* __restrict__ ws, float* __restrict__ out)
{
  __shared__ float sA[256], sB[256], sC[256];
  const int t = threadIdx.x;
  // NBLOCKS == 512 partials per statistic.
  sA[t] = ws[t]        + ws[t + 256];
  sB[t] = ws[512 + t]  + ws[512 + t + 256];
  sC[t] = ws[1024 + t] + ws[1024 + t + 256];
  __syncthreads();
  for (int off = 128; off > 0; off >>= 1) {
    if (t < off) {
      sA[t] += sA[t + off];
      sB[t] += sB[t + off];
      sC[t] += sC[t + off];
    }
    __syncthreads();
  }
  if (t == 0) {
    const float inter = sB[0];
    const float uni   = sA[0] + sC[0] - inter;
    out[0] = (inter + 1e-6f) / (uni + 1e-6f);
  }
}

extern "C" void kernel_launch(void* const* d_in, const int* in_sizes, int n_in,
                              void* d_out, int out_size, void* d_ws, size_t ws_size,
                              hipStream_t stream) {
  (void)in_sizes; (void)n_in; (void)out_size; (void)ws_size;
  const float* o1     = (const float*)d_in[0];
  const float* o2     = (const float*)d_in[1];
  const float* desire = (const float*)d_in[2];
  float* ws  = (float*)d_ws;   // 3 * NBLOCKS floats, fully rewritten every call
  float* out = (float*)d_out;

  dist_iou_partial_kernel<<<NBLOCKS, BLOCK, 0, stream>>>(o1, o2, desire, ws);
  dist_iou_finalize_kernel<<<1, 256, 0, stream>>>(ws, out);
}